// GCNLayerWithVirtualNode_30116310679887
// MI455X (gfx1250) — compile-verified
//
#include <hip/hip_runtime.h>

typedef __attribute__((ext_vector_type(2))) float v2f;
typedef __attribute__((ext_vector_type(8))) float v8f;

#define D 128
#define NGRAPH 256

// ---------------------------------------------------------------- copy H -> out
__global__ __launch_bounds__(256) void k_copy(const float* __restrict__ src,
                                              float* __restrict__ dst, int n4) {
  int i = blockIdx.x * blockDim.x + threadIdx.x;
  if (i < n4) ((float4*)dst)[i] = ((const float4*)src)[i];
}

// ---------------------------------------------------------------- zero scratch
__global__ __launch_bounds__(256) void k_zero(float* __restrict__ p, int n) {
  int i = blockIdx.x * blockDim.x + threadIdx.x;
  if (i < n) p[i] = 0.0f;
}

// ---------------------------------------------------------------- edge scatter:
// one wave32 per edge; lane handles 4 consecutive features (float4 gather,
// 4x global_atomic_add_f32 scatter).
__global__ __launch_bounds__(256) void k_edges(const int* __restrict__ ei,
                                               const float* __restrict__ H,
                                               float* __restrict__ out, int E) {
  int w = (blockIdx.x * blockDim.x + threadIdx.x) >> 5;
  int lane = threadIdx.x & 31;
  if (w >= E) return;
  int s = ei[w];
  int d = ei[E + w];
  const float4 v = *(const float4*)(H + (size_t)s * D + lane * 4);
  float* o = out + (size_t)d * D + lane * 4;
  atomicAdd(o + 0, v.x);
  atomicAdd(o + 1, v.y);
  atomicAdd(o + 2, v.z);
  atomicAdd(o + 3, v.w);
}

// ---------------------------------------------------------------- per-graph sums:
// batch is sorted, so a wave accumulates a contiguous run of 32 nodes in
// registers and only flushes atomics when the graph id changes.
__global__ __launch_bounds__(256) void k_gsum(const float* __restrict__ out,
                                              const int* __restrict__ batch,
                                              float* __restrict__ sums,
                                              float* __restrict__ counts, int N) {
  const int CHUNK = 32;
  int w = (blockIdx.x * blockDim.x + threadIdx.x) >> 5;
  int lane = threadIdx.x & 31;
  int n0 = w * CHUNK;
  if (n0 >= N) return;
  int n1 = min(n0 + CHUNK, N);
  float ax = 0.f, ay = 0.f, az = 0.f, aw = 0.f, cnt = 0.f;
  int cur = batch[n0];
  for (int n = n0; n < n1; ++n) {
    int g = batch[n];
    if (g != cur) {
      float* sp = sums + cur * D + lane * 4;
      atomicAdd(sp + 0, ax); atomicAdd(sp + 1, ay);
      atomicAdd(sp + 2, az); atomicAdd(sp + 3, aw);
      if (lane == 0) atomicAdd(counts + cur, cnt);
      ax = ay = az = aw = 0.f; cnt = 0.f; cur = g;
    }
    const float4 v = *(const float4*)(out + (size_t)n * D + lane * 4);
    ax += v.x; ay += v.y; az += v.z; aw += v.w;
    cnt += 1.f;
  }
  float* sp = sums + cur * D + lane * 4;
  atomicAdd(sp + 0, ax); atomicAdd(sp + 1, ay);
  atomicAdd(sp + 2, az); atomicAdd(sp + 3, aw);
  if (lane == 0) atomicAdd(counts + cur, cnt);
}

// ---------------------------------------------------------------- vmean = sums/count
__global__ __launch_bounds__(256) void k_vmean(const float* __restrict__ sums,
                                               const float* __restrict__ counts,
                                               float* __restrict__ vmean, int n) {
  int i = blockIdx.x * blockDim.x + threadIdx.x;
  if (i < n) {
    float c = counts[i >> 7];  // i / D
    vmean[i] = sums[i] / fmaxf(c, 1.0f);
  }
}

// ---------------------------------------------------------------- fused GEMM:
// out = relu( (out + vmean[batch]) @ W ), computed IN PLACE.
// Each wave owns a 16-row tile, loads its entire A (16x128 fp32, +vmean fused)
// into registers first, then accumulates 8 column tiles with
// v_wmma_f32_16x16x4_f32 (32 K-steps each) against W staged in LDS.
// LDS holds W pair-swizzled so one B fragment = one ds_load_b64 per lane:
//   Wl[((k>>1)*D + n)*2 + (k&1)]  (k = row of W, n = col of W)
__global__ __launch_bounds__(256) void k_gemm(float* __restrict__ out,
                                              const float* __restrict__ W,
                                              const int* __restrict__ batch,
                                              const float* __restrict__ vmean,
                                              int N) {
  __shared__ float Wl[D * D];  // 64 KB
  for (int i = threadIdx.x; i < D * D; i += 256) {
    int k = i >> 7, n = i & 127;
    Wl[(((k >> 1) * D + n) << 1) | (k & 1)] = W[i];
  }
  __syncthreads();

  int wave = threadIdx.x >> 5;
  int lane = threadIdx.x & 31;
  int row0 = (blockIdx.x * 8 + wave) * 16;
  if (row0 >= N) return;  // wave-uniform: EXEC stays all-ones for WMMA

  // A fragment addressing (16x4 fp32 tile layout):
  //   lanes 0-15 : M = lane,    VGPR0 = K+0, VGPR1 = K+1
  //   lanes 16-31: M = lane-16, VGPR0 = K+2, VGPR1 = K+3
  int mrow = row0 + (lane & 15);
  int koff = (lane < 16) ? 0 : 2;
  int g = batch[mrow];

  const float* arow = out + (size_t)mrow * D;
  const float* vrow = vmean + g * D;
  v2f a[32];
#pragma unroll
  for (int ks = 0; ks < 32; ++ks) {
    int k = ks * 4 + koff;
    v2f h  = *(const v2f*)(arow + k);
    v2f vm = *(const v2f*)(vrow + k);
    a[ks] = h + vm;  // fused virtual-node broadcast
  }

  const v2f* Wl2 = (const v2f*)Wl;
  int ncol = lane & 15;
  int rbase = row0 + ((lane < 16) ? 0 : 8);  // C/D layout: VGPR v -> M = v (+8 hi lanes)

  for (int ct = 0; ct < 8; ++ct) {
    v8f c = {};
#pragma unroll
    for (int ks = 0; ks < 32; ++ks) {
      int kb = ks * 4 + koff;  // even -> one b64 LDS load gives (K, K+1) pair
      v2f b = Wl2[(kb >> 1) * D + ct * 16 + ncol];
      c = __builtin_amdgcn_wmma_f32_16x16x4_f32(false, a[ks], false, b,
                                                (short)0, c, false, false);
    }
    int col = ct * 16 + ncol;
#pragma unroll
    for (int v = 0; v < 8; ++v)
      out[(size_t)(rbase + v) * D + col] = fmaxf(c[v], 0.0f);
  }
}

// ----------------------------------------------------------------
extern "C" void kernel_launch(void* const* d_in, const int* in_sizes, int n_in,
                              void* d_out, int out_size, void* d_ws, size_t ws_size,
                              hipStream_t stream) {
  const float* H   = (const float*)d_in[0];
  const int* ei    = (const int*)d_in[1];
  const int* batch = (const int*)d_in[2];
  const float* W   = (const float*)d_in[3];
  float* out = (float*)d_out;

  int N = in_sizes[0] / D;  // 100000
  int E = in_sizes[1] / 2;  // 1600000

  float* sums   = (float*)d_ws;        // NGRAPH*D
  float* counts = sums + NGRAPH * D;   // NGRAPH
  float* vmean  = counts + NGRAPH;     // NGRAPH*D

  int n4 = N * D / 4;
  k_copy<<<(n4 + 255) / 256, 256, 0, stream>>>(H, out, n4);

  int nz = NGRAPH * D + NGRAPH;
  k_zero<<<(nz + 255) / 256, 256, 0, stream>>>(sums, nz);

  long long ethreads = (long long)E * 32;
  k_edges<<<(int)((ethreads + 255) / 256), 256, 0, stream>>>(ei, H, out, E);

  int nwarps = (N + 31) / 32;
  k_gsum<<<(nwarps * 32 + 255) / 256, 256, 0, stream>>>(out, batch, sums, counts, N);

  int nv = NGRAPH * D;
  k_vmean<<<(nv + 255) / 256, 256, 0, stream>>>(sums, counts, vmean, nv);

  int ntiles = (N + 15) / 16;
  k_gemm<<<(ntiles + 7) / 8, 256, 0, stream>>>(out, W, batch, vmean, N);
}